// EmbeddingsDecoder_9079560864583
// MI455X (gfx1250) — compile-verified
//
#include <hip/hip_runtime.h>
#include <math.h>

typedef float v2f __attribute__((ext_vector_type(2)));
typedef float v8f __attribute__((ext_vector_type(8)));

#define BDIM 256
#define KC   256
#define LDSP (KC + 4)   // padded LDS row stride (floats); 1040B rows keep b128 dsts 16B-aligned

// ---- CDNA5 async-to-LDS path (guarded; falls back to sync staging) --------
#if defined(__has_builtin)
#  if __has_builtin(__builtin_amdgcn_global_load_async_to_lds_b128)
#    define HAVE_ASYNC_LDS 1
#  endif
#endif
#ifndef HAVE_ASYNC_LDS
#  define HAVE_ASYNC_LDS 0
#endif

#if HAVE_ASYNC_LDS
// Builtin signature (from hipcc diagnostic): param0 is a 16-byte int-vector
// pointer in addrspace(1) (printed as "int __vector(4) __device__ *").
typedef int b128_t __attribute__((vector_size(16)));
typedef __attribute__((address_space(1))) b128_t* gb128p;   // global
typedef __attribute__((address_space(3))) b128_t* lb128p;   // LDS
#  if defined(__has_builtin)
#    if __has_builtin(__builtin_amdgcn_s_wait_asynccnt)
#      define WAIT_ASYNC(n) __builtin_amdgcn_s_wait_asynccnt((unsigned short)(n))
#    endif
#  endif
#  ifndef WAIT_ASYNC
#    define WAIT_ASYNC(n) asm volatile("s_wait_asynccnt %0" ::"i"(n) : "memory")
#  endif
#else
#  define WAIT_ASYNC(n) ((void)0)
#endif

// Stage X[16 rows x KC cols] starting at column kc into LDS tile dstBase.
// Xm is pre-offset to row m0. 1024 x b128 transfers, 4 per thread, coalesced.
__device__ __forceinline__ void stage_tile(const float* __restrict__ Xm, int K,
                                           int kc, float* dstBase, int tid) {
#if HAVE_ASYNC_LDS
    #pragma unroll
    for (int q = 0; q < 4; ++q) {
        int idx = tid + q * BDIM;          // 0..1023
        int row = idx >> 6;                // KC/4 = 64 float4 per row
        int c4  = idx & 63;
        const float* src = Xm + (size_t)row * (size_t)K + kc + c4 * 4;
        float* dst = dstBase + row * LDSP + c4 * 4;
        __builtin_amdgcn_global_load_async_to_lds_b128(
            (gb128p)const_cast<float*>(src),
            (lb128p)dst,
            /*imm offset*/0, /*cpol*/0);
    }
#else
    #pragma unroll
    for (int q = 0; q < 4; ++q) {
        int idx = tid + q * BDIM;
        int row = idx >> 6;
        int c4  = idx & 63;
        float4 v = reinterpret_cast<const float4*>(Xm + (size_t)row * (size_t)K + kc)[c4];
        float* dst = dstBase + row * LDSP + c4 * 4;
        dst[0] = v.x; dst[1] = v.y; dst[2] = v.z; dst[3] = v.w;
    }
#endif
}

// ---------------------------------------------------------------------------
// Kernel 1: logits = X[M,K] * W[V,K]^T  via V_WMMA_F32_16X16X4_F32
// Block: 256 thr = 8 waves. Block tile M=16, N=512; each wave: 16x64 (4 accs).
// X tile double-buffered in LDS via async-to-LDS; W streamed from global.
// ---------------------------------------------------------------------------
__global__ __launch_bounds__(BDIM) void gemm_logits_wmma_f32(
    const float* __restrict__ X,     // [M, K]
    const float* __restrict__ W,     // [V, K]
    float* __restrict__ logits,      // [M, V]
    int M, int K, int V)
{
    __shared__ float sX[2][16 * LDSP];

    const int tid  = threadIdx.x;
    const int lane = tid & 31;
    const int wave = tid >> 5;       // 0..7
    const int half = lane >> 4;      // 0/1
    const int l16  = lane & 15;

    const int m0 = blockIdx.x * 16;
    const int n0 = blockIdx.y * 512 + wave * 64;

    // Clamp W rows for the N tail; OOB columns are discarded at store time.
    const float* Wrow[4];
    #pragma unroll
    for (int t = 0; t < 4; ++t) {
        int vr = n0 + 16 * t + l16;
        if (vr >= V) vr = V - 1;
        Wrow[t] = W + (size_t)vr * (size_t)K;
    }

    const float* Xm = X + (size_t)m0 * (size_t)K;

    v8f acc[4] = {};
    const int nChunks = K / KC;      // K = 1024 -> 4

    stage_tile(Xm, K, 0, &sX[0][0], tid);

    for (int ci = 0; ci < nChunks; ++ci) {
        const int kc = ci * KC;
        if (ci + 1 < nChunks) {
            // Prefetch next chunk into the other buffer (safe: barrier at end of
            // previous iteration guarantees no wave still reads that buffer).
            stage_tile(Xm, K, kc + KC, &sX[(ci + 1) & 1][0], tid);
            WAIT_ASYNC(4);           // async loads complete in order: current 4 done
            #pragma unroll
            for (int t = 0; t < 4; ++t)
                __builtin_prefetch(Wrow[t] + kc + KC, 0, 3);
        } else {
            WAIT_ASYNC(0);
        }
        __syncthreads();

        const float* aBase = &sX[ci & 1][l16 * LDSP];

        // A layout: lanes 0-15 hold {K=0,1}, lanes 16-31 hold {K=2,3}, M = lane%16
        // B layout: lanes 0-15 hold {K=0,1}, lanes 16-31 hold {K=2,3}, N = lane%16
        #pragma unroll 4
        for (int kk = 0; kk < KC; kk += 4) {
            const int ko = kk + 2 * half;
            v2f a; a.x = aBase[ko]; a.y = aBase[ko + 1];
            #pragma unroll
            for (int t = 0; t < 4; ++t) {
                v2f b; b.x = Wrow[t][kc + ko]; b.y = Wrow[t][kc + ko + 1];
                acc[t] = __builtin_amdgcn_wmma_f32_16x16x4_f32(
                             /*neg_a=*/false, a, /*neg_b=*/false, b,
                             /*c_mod=*/(short)0, acc[t],
                             /*reuse_a=*/false, /*reuse_b=*/false);
            }
        }
        __syncthreads();
    }

    // C/D layout: VGPR r -> (M=r, N=lane) lanes 0-15 ; (M=r+8, N=lane-16) lanes 16-31
    #pragma unroll
    for (int t = 0; t < 4; ++t) {
        const int gn = n0 + 16 * t + l16;
        if (gn < V) {
            float* out = logits + (size_t)(m0 + 8 * half) * (size_t)V + gn;
            #pragma unroll
            for (int r = 0; r < 8; ++r) {
                out[0] = acc[t][r];
                out += V;
            }
        }
    }
}

// ---------------------------------------------------------------------------
// Kernel 2: per-row max and sum(exp(l - max)) -> stats[row] = {max, sum}
// One block per row; second pass over the row hits L2 (row = 200KB << 192MB).
// ---------------------------------------------------------------------------
__global__ __launch_bounds__(BDIM) void softmax_stats(
    const float* __restrict__ logits, float* __restrict__ stats, int V)
{
    __shared__ float red[BDIM];
    const int row = blockIdx.x;
    const int tid = threadIdx.x;
    const float* __restrict__ l = logits + (size_t)row * (size_t)V;

    float m = -INFINITY;
    for (int v = tid; v < V; v += BDIM) m = fmaxf(m, l[v]);
    red[tid] = m;
    __syncthreads();
    #pragma unroll
    for (int s = BDIM / 2; s > 0; s >>= 1) {
        if (tid < s) red[tid] = fmaxf(red[tid], red[tid + s]);
        __syncthreads();
    }
    const float rowmax = red[0];
    __syncthreads();

    float sum = 0.0f;
    for (int v = tid; v < V; v += BDIM) sum += __expf(l[v] - rowmax);
    red[tid] = sum;
    __syncthreads();
    #pragma unroll
    for (int s = BDIM / 2; s > 0; s >>= 1) {
        if (tid < s) red[tid] += red[tid + s];
        __syncthreads();
    }
    if (tid == 0) {
        stats[row * 2 + 0] = rowmax;
        stats[row * 2 + 1] = red[0];
    }
}

// ---------------------------------------------------------------------------
// Kernel 3: probs = exp(logits - max) / sum   (row given by blockIdx.y)
// ---------------------------------------------------------------------------
__global__ __launch_bounds__(BDIM) void softmax_norm(
    const float* __restrict__ logits, const float* __restrict__ stats,
    float* __restrict__ probs, int V)
{
    const int row = blockIdx.y;
    const int v   = blockIdx.x * BDIM + threadIdx.x;
    if (v < V) {
        const float m  = stats[row * 2 + 0];
        const float rs = 1.0f / stats[row * 2 + 1];
        const size_t i = (size_t)row * (size_t)V + v;
        probs[i] = __expf(logits[i] - m) * rs;
    }
}

// ---------------------------------------------------------------------------
extern "C" void kernel_launch(void* const* d_in, const int* in_sizes, int n_in,
                              void* d_out, int out_size, void* d_ws, size_t ws_size,
                              hipStream_t stream) {
    (void)in_sizes; (void)n_in; (void)out_size; (void)ws_size;

    constexpr int B = 2, S = 2048, D = 1024, V = 50257;
    constexpr int M = B * S;                       // 4096 rows

    const float* x = (const float*)d_in[0];        // [B,S,D] fp32
    const float* w = (const float*)d_in[1];        // [V,D]   fp32

    float* probs  = (float*)d_out;                 // output 0: [B,S,V]
    float* logits = probs + (size_t)M * (size_t)V; // output 1: [B,S,V]
    float* stats  = (float*)d_ws;                  // [M][2] = {max, sumexp}

    // 1) logits = x @ w^T  (WMMA f32, async double-buffered X staging)
    dim3 g1(M / 16, (V + 511) / 512);
    gemm_logits_wmma_f32<<<g1, BDIM, 0, stream>>>(x, w, logits, M, D, V);

    // 2) per-row softmax stats
    softmax_stats<<<M, BDIM, 0, stream>>>(logits, stats, V);

    // 3) normalize
    dim3 g3((V + BDIM - 1) / BDIM, M);
    softmax_norm<<<g3, BDIM, 0, stream>>>(logits, stats, probs, V);
}